// GaussianShape_85229331022485
// MI455X (gfx1250) — compile-verified
//
#include <hip/hip_runtime.h>
#include <hip/hip_bf16.h>

// ---------------------------------------------------------------------------
// GaussianShape filterbank apply:  out[b,n,t] = sum_f fb[n,f] * in[b,f,t]
//   B=64, N=64 filters, F=K=257, T=3000, fp32 throughout.
// fb is banded (<=9 nonzeros/row); filters sorted by center -> each group of
// 16 filters covers a contiguous K window. FP32 WMMA (16x16x4) runs only over
// each group's [klo,khi) window (~282 K-steps total vs 4*257 dense), keeping
// exact fp32 accumulation while staying under the HBM roofline
// (~246 MB mandatory traffic @ 23.3 TB/s ~= 10.6 us).
// ---------------------------------------------------------------------------

#define NFILT 64
#define FSZ   257            // FILTER_SIZE == NFFT/2+1
#define KPAD  260            // FSZ padded up to multiple of 4 (zero rows)
#define TDIM  3000
#define BDIM  64
#define HALF  4

typedef __attribute__((ext_vector_type(2))) float v2f;
typedef __attribute__((ext_vector_type(8))) float v8f;

// --- Kernel 1: build fbanks [64][KPAD] (sorted, banded gaussians) + per-group
// --- K ranges [klo,khi) aligned to 4. One block of 64 threads, one-time cost.
__global__ void gs_build_fbanks(const float* __restrict__ w,
                                float* __restrict__ fb,
                                int* __restrict__ kranges) {
    __shared__ float s_cn[NFILT], s_sg[NFILT];
    __shared__ float o_cn[NFILT], o_sg[NFILT];
    __shared__ int   s_lo[NFILT], s_hi[NFILT];

    const int i = threadIdx.x;              // 0..63, one filter per thread
    float cn = w[2 * i];                    // w[:,0]
    float sg = w[2 * i + 1];                // w[:,1]
    sg = fminf(fmaxf(sg, 1e-5f), 0.5f);     // clip(sigma, 1e-5, 0.5)
    cn = fminf(fmaxf(cn, 30.0f / 80000.0f), 1.0f);
    s_cn[i] = cn;
    s_sg[i] = sg;
    __syncthreads();

    // stable argsort by cn: rank = #(smaller) + #(equal with lower index)
    int r = 0;
    for (int j = 0; j < NFILT; ++j) {
        float cj = s_cn[j];
        r += (cj < cn) || (cj == cn && j < i) ? 1 : 0;
    }
    o_cn[r] = cn;
    o_sg[r] = sg;
    __syncthreads();

    // thread i now owns sorted filter row i
    const float c = o_cn[i];
    const float s = o_sg[i];
    const int ci = (int)floorf(c * (float)FSZ);
    const int lo = (ci - HALF > 0) ? (ci - HALF) : 0;
    const int hi = (ci + HALF < FSZ) ? (ci + HALF + 1) : (FSZ - 1);
    s_lo[i] = lo;
    s_hi[i] = hi;

    const float inv_s = 1.0f / s;
    for (int j = 0; j < KPAD; ++j) {
        float v = 0.0f;
        if (j >= lo && j < hi) {
            float z = ((float)j / (float)FSZ - c) * inv_s;
            v = expf(-0.5f * z * z);
        }
        fb[i * KPAD + j] = v;
    }
    __syncthreads();

    // per 16-filter group: union K window, aligned to WMMA K=4
    if (i < 4) {
        int klo = KPAD, khi = 0;
        for (int j = 0; j < 16; ++j) {
            int lj = s_lo[i * 16 + j], hj = s_hi[i * 16 + j];
            klo = (lj < klo) ? lj : klo;
            khi = (hj > khi) ? hj : khi;
        }
        klo &= ~3;
        khi = (khi + 3) & ~3;                // <= 260 (KPAD); may read input rows >= FSZ
        kranges[2 * i]     = klo;
        kranges[2 * i + 1] = khi;
    }
}

// --- GEMM core. FULL=true: whole 64-wide t-tile in range -> zero predication,
// --- compile-time load/store offsets. FULL=false: clamp OOB lanes' column to
// --- TDIM-1 for loads (garbage -> never-stored accumulator columns), guard
// --- only the stores.
template <bool FULL>
__device__ __forceinline__ void gs_gemm_core(const float* __restrict__ inB,
                                             const float* __restrict__ fbA,
                                             float* __restrict__ outp,
                                             int klo, int khi, int kh, int tbase) {
    int tc[4];
#pragma unroll
    for (int tt = 0; tt < 4; ++tt) {
        if (FULL) {
            tc[tt] = tt * 16;                          // compile-time constant
        } else {
            int t = tbase + tt * 16;
            tc[tt] = (t < TDIM ? t : (TDIM - 1)) - tbase;   // clamped offset
        }
    }

    v8f acc[4] = {};
    for (int k = klo; k < khi; k += 4) {
        v2f a;
        a.x = fbA[k + kh];
        a.y = fbA[k + kh + 1];
        const float* p0 = inB + (size_t)(k + kh) * TDIM + tbase;
        const float* p1 = p0 + TDIM;
#pragma unroll
        for (int tt = 0; tt < 4; ++tt) {
            v2f bv;
            bv.x = p0[tc[tt]];
            bv.y = p1[tc[tt]];
            acc[tt] = __builtin_amdgcn_wmma_f32_16x16x4_f32(
                false, a, false, bv, (short)0, acc[tt], false, false);
        }
    }

#pragma unroll
    for (int tt = 0; tt < 4; ++tt) {
        int t = tbase + tt * 16;
        if (FULL || t < TDIM) {
#pragma unroll
            for (int v = 0; v < 8; ++v) {
                outp[(size_t)v * TDIM + tt * 16] = acc[tt][v];
            }
        }
    }
}

// --- Kernel 2: banded GEMM via V_WMMA_F32_16X16X4_F32.
// Block = 128 threads (4 waves). Wave g handles filter group g (rows 16g..16g+15)
// and a 16(n) x 64(t) output tile: 4 independent accumulators per wave.
//
// F32 WMMA operand layouts (ISA 7.12.2):
//   A 16x4 : lanes 0-15 -> M=lane, VGPR0=K0, VGPR1=K1 ; lanes 16-31 -> K2,K3
//   B 4x16 : lanes 0-15 -> N=lane, VGPR0=K0, VGPR1=K1 ; lanes 16-31 -> K2,K3
//   C 16x16: VGPR v -> M = v + 8*(lane>=16), N = lane%16
__global__ void gs_banded_gemm(const float* __restrict__ in,
                               const float* __restrict__ fb,
                               const int* __restrict__ kranges,
                               float* __restrict__ out) {
    const int lane = threadIdx.x & 31;
    const int g    = threadIdx.x >> 5;        // wave id == filter group 0..3
    const int b    = blockIdx.y;              // batch
    const int t0   = blockIdx.x * 64;         // 64-wide t tile per block

    const int m  = lane & 15;                 // M (A rows) / N (B cols) index
    const int kh = (lane >> 4) << 1;          // 0 for lanes 0-15, 2 for 16-31

    const int klo = __builtin_amdgcn_readfirstlane(kranges[2 * g]);
    const int khi = __builtin_amdgcn_readfirstlane(kranges[2 * g + 1]);

    const int n0 = g * 16;
    const int rbase = (lane < 16) ? 0 : 8;
    const int tbase = t0 + m;

    const float* __restrict__ fbA = fb + (size_t)(n0 + m) * KPAD;
    const float* __restrict__ inB = in + (size_t)b * FSZ * TDIM;
    float* __restrict__ outp =
        out + ((size_t)b * NFILT + n0 + rbase) * TDIM + tbase;

    if (t0 + 64 <= TDIM) {   // uniform across the block -> scalar branch
        gs_gemm_core<true>(inB, fbA, outp, klo, khi, kh, tbase);
    } else {
        gs_gemm_core<false>(inB, fbA, outp, klo, khi, kh, tbase);
    }
}

extern "C" void kernel_launch(void* const* d_in, const int* in_sizes, int n_in,
                              void* d_out, int out_size, void* d_ws, size_t ws_size,
                              hipStream_t stream) {
    (void)in_sizes; (void)n_in; (void)out_size; (void)ws_size;
    const float* inputs = (const float*)d_in[0];   // [64, 257, 3000] fp32
    const float* w      = (const float*)d_in[1];   // [64, 2] fp32
    float* out          = (float*)d_out;           // [64, 64, 3000] fp32

    float* fb   = (float*)d_ws;                                      // 64*260 f32
    int*   krng = (int*)((char*)d_ws + NFILT * KPAD * sizeof(float)); // 8 ints

    gs_build_fbanks<<<1, NFILT, 0, stream>>>(w, fb, krng);

    dim3 grid((TDIM + 63) / 64, BDIM, 1);   // 47 t-tiles x 64 batches
    gs_banded_gemm<<<grid, 128, 0, stream>>>(inputs, fb, krng, out);
}